// Quanv2d_80805514706995
// MI455X (gfx1250) — compile-verified
//
#include <hip/hip_runtime.h>

typedef __attribute__((ext_vector_type(2))) float v2f;
typedef __attribute__((ext_vector_type(8))) float v8f;

#define DEVINL __device__ __forceinline__

// Apply real 2x2 gate [[m00,m01],[m10,m11]] on the qubit selected by MASK
// (qubit q <-> MASK = 8 >> q, since qubit 0 is the MSB of the state index).
template<int MASK>
DEVINL void mix2(float* re, float* im, float m00, float m01, float m10, float m11) {
#pragma unroll
    for (int s = 0; s < 16; ++s) {
        if ((s & MASK) == 0) {
            const int t = s | MASK;
            const float r0 = re[s], r1 = re[t];
            re[s] = m00 * r0 + m01 * r1;
            re[t] = m10 * r0 + m11 * r1;
            const float i0 = im[s], i1 = im[t];
            im[s] = m00 * i0 + m01 * i1;
            im[t] = m10 * i0 + m11 * i1;
        }
    }
}

// z[s] *= exp(i * 0.5*theta * sign), sign = +1 if bit set else -1
template<int MASK>
DEVINL void rz(float* re, float* im, float theta) {
    float sn, cs;
    __sincosf(0.5f * theta, &sn, &cs);
#pragma unroll
    for (int s = 0; s < 16; ++s) {
        const float sg = (s & MASK) ? sn : -sn;
        const float r = re[s], i = im[s];
        re[s] = cs * r - sg * i;
        im[s] = sg * r + cs * i;
    }
}

template<int MASK>
DEVINL void ry(float* re, float* im, float theta) {
    float sn, cs;
    __sincosf(0.5f * theta, &sn, &cs);
    mix2<MASK>(re, im, cs, -sn, sn, cs);
}

// Controlled phase: states with control bit 1 get exp(i*0.5*w*(2*bt-1))
template<int CMASK, int TMASK>
DEVINL void crz(float* re, float* im, float w) {
    float sn, cs;
    __sincosf(0.5f * w, &sn, &cs);
#pragma unroll
    for (int s = 0; s < 16; ++s) {
        if (s & CMASK) {
            const float sg = (s & TMASK) ? sn : -sn;
            const float r = re[s], i = im[s];
            re[s] = cs * r - sg * i;
            im[s] = sg * r + cs * i;
        }
    }
}

// One statevector per lane (wave32). Epilogue: probs @ Z via v_wmma_f32_16x16x4_f32.
__global__ __launch_bounds__(256) void quanv2d_kernel(
        const float* __restrict__ X,      // [B,1,256,256]
        const float* __restrict__ W,      // [1,8]
        float* __restrict__ out,          // [N,4] flat
        int Ntotal) {
    constexpr int P  = 127 * 127;  // patches per image
    constexpr int HW = 256 * 256;

    const int lane = threadIdx.x & 31;
    const int wv   = threadIdx.x >> 5;
    const int n    = blockIdx.x * 256 + threadIdx.x;   // statevector index
    const int nc   = (n < Ntotal) ? n : 0;             // clamp for tail lanes

    // ---- decode patch coordinates and load 16 angles ----
    const int b  = nc / P;
    const int p  = nc - b * P;
    const int pi = p / 127;
    const int pj = p - pi * 127;
    const float* src = X + (size_t)b * HW + (size_t)(pi * 2) * 256 + (pj * 2);

    float x[16];
#pragma unroll
    for (int di = 0; di < 4; ++di)
#pragma unroll
        for (int dj = 0; dj < 4; ++dj)
            x[di * 4 + dj] = src[di * 256 + dj];

    float w[8];
#pragma unroll
    for (int i = 0; i < 8; ++i) w[i] = W[i];

    // ---- statevector simulation, fully in registers ----
    float re[16], im[16];
#pragma unroll
    for (int s = 0; s < 16; ++s) { re[s] = 0.0f; im[s] = 0.0f; }
    re[0] = 1.0f;

    const float inv = 0.70710678118654752440f;

    // qubit 0 (mask 8): H, then RZ/RY/RZ/RY/RZ with x[0..4]
    mix2<8>(re, im, inv, inv, inv, -inv);
    rz<8>(re, im, x[0]);  ry<8>(re, im, x[1]);
    rz<8>(re, im, x[2]);  ry<8>(re, im, x[3]);
    rz<8>(re, im, x[4]);
    // qubit 1 (mask 4): x[5..9]
    mix2<4>(re, im, inv, inv, inv, -inv);
    rz<4>(re, im, x[5]);  ry<4>(re, im, x[6]);
    rz<4>(re, im, x[7]);  ry<4>(re, im, x[8]);
    rz<4>(re, im, x[9]);
    // qubit 2 (mask 2): x[10..14]
    mix2<2>(re, im, inv, inv, inv, -inv);
    rz<2>(re, im, x[10]); ry<2>(re, im, x[11]);
    rz<2>(re, im, x[12]); ry<2>(re, im, x[13]);
    rz<2>(re, im, x[14]);
    // qubit 3 (mask 1): x[15] only
    mix2<1>(re, im, inv, inv, inv, -inv);
    rz<1>(re, im, x[15]);

    // entangling layer: CRZ ring, then RY per qubit
    crz<8, 4>(re, im, w[0]);
    crz<4, 2>(re, im, w[1]);
    crz<2, 1>(re, im, w[2]);
    crz<1, 8>(re, im, w[3]);
    ry<8>(re, im, w[4]);
    ry<4>(re, im, w[5]);
    ry<2>(re, im, w[6]);
    ry<1>(re, im, w[7]);

    // ---- stage probabilities to LDS for the WMMA epilogue ----
    __shared__ float probs[8][32][17];   // pad to 17 -> no 16-stride bank conflicts
#pragma unroll
    for (int s = 0; s < 16; ++s)
        probs[wv][lane][s] = re[s] * re[s] + im[s] * im[s];
    __syncthreads();

    // ---- E^T[qubit, sv] = Z^T (16x16, rows>=4 zero) x probs^T via 4 chained
    //      v_wmma_f32_16x16x4_f32 per group of 16 statevectors ----
    const int half = lane >> 4;          // 0: K rows {0,1}; 1: K rows {2,3}
    const int col  = lane & 15;          // A: M index (qubit); B: N index (sv)
    const int waveBase = blockIdx.x * 256 + wv * 32;
    const int sh = (col < 4) ? (3 - col) : 0;

#pragma unroll
    for (int g = 0; g < 2; ++g) {
        v8f acc = {};
#pragma unroll
        for (int k = 0; k < 4; ++k) {
            const int kk0 = 4 * k + 2 * half;   // this lane's first K row
            // A fragment: A[m=col][kk] = Z[kk][col] = 1-2*bit(kk, col) for col<4, else 0
            const float a0 = (col < 4) ? (1.0f - 2.0f * (float)((kk0 >> sh) & 1)) : 0.0f;
            const float a1 = (col < 4) ? (1.0f - 2.0f * (float)(((kk0 + 1) >> sh) & 1)) : 0.0f;
            v2f afrag = {a0, a1};
            // B fragment: B[kk][n=col] = probs[sv g*16+col][kk]
            v2f bfrag = {probs[wv][g * 16 + col][kk0],
                         probs[wv][g * 16 + col][kk0 + 1]};
            acc = __builtin_amdgcn_wmma_f32_16x16x4_f32(
                false, afrag, false, bfrag, (short)0, acc, false, false);
        }
        // D rows 0..3 on lanes 0..15 hold E[sv, qubit 0..3] -> one float4 store
        const long sv = (long)waveBase + g * 16 + col;
        if (half == 0 && sv < (long)Ntotal) {
            float4 r4 = make_float4(acc[0], acc[1], acc[2], acc[3]);
            *reinterpret_cast<float4*>(out + sv * 4) = r4;
        }
    }
}

extern "C" void kernel_launch(void* const* d_in, const int* in_sizes, int n_in,
                              void* d_out, int out_size, void* d_ws, size_t ws_size,
                              hipStream_t stream) {
    const float* X = (const float*)d_in[0];   // [B,1,256,256] f32
    const float* W = (const float*)d_in[1];   // [1,8] f32
    float* out = (float*)d_out;               // [B, P*4] f32

    const int B      = in_sizes[0] / (256 * 256);
    const int Ntotal = B * 127 * 127;         // statevectors
    const int blocks = (Ntotal + 255) / 256;

    quanv2d_kernel<<<blocks, 256, 0, stream>>>(X, W, out, Ntotal);
}